// SVFNetMap_88381837017173
// MI455X (gfx1250) — compile-verified
//
#include <hip/hip_runtime.h>
#include <stdint.h>

// ---------------------------------------------------------------------------
// Fused RK4 advection step for MI455X (gfx1250).
//   dphi/dt = -(v . grad) phi,  np.gradient-style differences, 10 RK4 steps.
// One kernel launch per RK4 step; all 4 stages fused via LDS tiling with a
// radius-4 halo.  Interior tiles are DMA'd into LDS with the CDNA5 Tensor
// Data Mover (tensor_load_to_lds, TENSORcnt).  Requires ws_size >= 128 MB.
// ---------------------------------------------------------------------------

#define SZ      4096
#define TILE    64              // tile side incl. halo (cells)
#define HALO    4               // RK4 = 4 nested radius-1 stencils
#define INTR    56              // TILE - 2*HALO
#define NB      74              // ceil(SZ / INTR)
#define NTHR    256             // 8 waves (wave32)
#define DT_F    0.1f
#define INV_H   4095.0f         // 1/h,   h = 1/(SZ-1)
#define INV_2H  2047.5f         // 1/(2h)

typedef unsigned int u32x4 __attribute__((ext_vector_type(4)));
typedef int          i32x8 __attribute__((ext_vector_type(8)));
typedef int          i32x4 __attribute__((ext_vector_type(4)));
typedef float        v2f   __attribute__((ext_vector_type(2)));
typedef float        v8f   __attribute__((ext_vector_type(8)));

__device__ __forceinline__ int iclampi(int v, int lo, int hi) {
  return v < lo ? lo : (v > hi ? hi : v);
}

// RHS of dphi/dt = -(v.grad)phi at one cell, np.gradient boundary rules.
// f, vv: [TILE][TILE] float2 tiles in LDS (both map components per cell).
// (gy,gx) may lie outside the domain for junk halo cells of boundary tiles;
// clamped indexing keeps all LDS reads in-bounds and junk is never consumed.
__device__ __forceinline__ float2 rhs_cell(const float2* __restrict__ f,
                                           const float2* __restrict__ vv,
                                           int ly, int lx, int gy, int gx) {
  const int gyc = iclampi(gy, 0, SZ - 1);
  const int gxc = iclampi(gx, 0, SZ - 1);
  const int oym = (gyc > 0)      ? -1 : 0;
  const int oyp = (gyc < SZ - 1) ?  1 : 0;
  const int oxm = (gxc > 0)      ? -1 : 0;
  const int oxp = (gxc < SZ - 1) ?  1 : 0;
  const float wy = ((oyp - oym) == 2) ? INV_2H : INV_H;  // one-sided at edges
  const float wx = ((oxp - oxm) == 2) ? INV_2H : INV_H;
  const float2 yp = f[(ly + oyp) * TILE + lx];
  const float2 ym = f[(ly + oym) * TILE + lx];
  const float2 xp = f[ly * TILE + (lx + oxp)];
  const float2 xm = f[ly * TILE + (lx + oxm)];
  const float d0x = (yp.x - ym.x) * wy;   // d/d(axis0) of comp 0
  const float d0y = (yp.y - ym.y) * wy;   // d/d(axis0) of comp 1
  const float d1x = (xp.x - xm.x) * wx;   // d/d(axis1) of comp 0
  const float d1y = (xp.y - xm.y) * wx;   // d/d(axis1) of comp 1
  const float2 vc = vv[ly * TILE + lx];
  float2 r;
  r.x = -(vc.x * d0x + vc.y * d1x);
  r.y = -(vc.x * d0y + vc.y * d1y);
  return r;
}

// dst = p + coef * rhs(src) over the shrinking region [lo, TILE-lo)^2.
__device__ __forceinline__ void stage(float2* __restrict__ dst,
                                      const float2* __restrict__ src,
                                      const float2* __restrict__ sp,
                                      const float2* __restrict__ sv,
                                      float coef, int lo,
                                      int y0, int x0, int tid) {
  const int side = TILE - 2 * lo;
  for (int i = tid; i < side * side; i += NTHR) {
    const int ly = lo + i / side;
    const int lx = lo + i % side;
    const float2 r = rhs_cell(src, sv, ly, lx, y0 + ly, x0 + lx);
    const float2 p = sp[ly * TILE + lx];
    dst[ly * TILE + lx] = make_float2(p.x + coef * r.x, p.y + coef * r.y);
  }
}

#if __has_builtin(__builtin_amdgcn_tensor_load_to_lds) && defined(__gfx1250__)
#define HAVE_TDM 1
// Issue one TDM descriptor: 2D tile (TILE rows x TILE*2 floats) from global
// row-major tensor [SZ][SZ*2] f32 into contiguous LDS.  D# bit layout per
// CDNA5 ISA ch.8 (group0: count/lds/global/type, group1: dims/tile/strides).
// This toolchain exposes the 6-arg builtin (clang-23 / therock form):
//   (u32x4 g0, i32x8 g1, i32x4 g2, i32x4 g3, i32x8 extra, i32 cpol)
__device__ __forceinline__ void tdm_load_tile(const float2* gbase, int y0, int x0,
                                              float2* lds) {
  const unsigned long long gaddr =
      (unsigned long long)(uintptr_t)gbase +
      (unsigned long long)((long long)y0 * SZ + x0) * 8ull;
  const unsigned int ldsa = (unsigned int)(uintptr_t)lds;  // low 32 = LDS offset
  u32x4 g0;
  g0.x = 1u;                                              // count=1, user mode
  g0.y = ldsa;                                            // lds_addr (bytes)
  g0.z = (unsigned int)(gaddr & 0xffffffffu);             // global_addr[31:0]
  g0.w = (unsigned int)((gaddr >> 32) & 0x01ffffffu)      // global_addr[56:32]
         | (2u << 30);                                    // type = 2 ("image")
  i32x8 g1;
  g1[0] = (int)(2u << 16);                                // data_size = 4 bytes
  g1[1] = (int)(((unsigned)(SZ * 2) & 0xffffu) << 16);    // tensor_dim0 lo16
  g1[2] = (int)((((unsigned)(SZ * 2)) >> 16) |            // tensor_dim0 hi16
                (((unsigned)SZ & 0xffffu) << 16));        // tensor_dim1 lo16
  g1[3] = (int)((((unsigned)SZ) >> 16) |                  // tensor_dim1 hi16
                (((unsigned)(TILE * 2)) << 16));          // tile_dim0 = 128
  g1[4] = (int)TILE;                                      // tile_dim1=64, dim2=0
  g1[5] = (int)(SZ * 2);                                  // dim0_stride lo32
  g1[6] = 0;                                              // dim0_stride hi16 | dim1_stride lo16
  g1[7] = 0;                                              // dim1_stride hi32
  const i32x4 z4 = {0, 0, 0, 0};                          // groups 2/3 unused (2D)
  const i32x8 z8 = {0, 0, 0, 0, 0, 0, 0, 0};              // extra group (zero)
  __builtin_amdgcn_tensor_load_to_lds(g0, g1, z4, z4, z8, 0);
}
#else
#define HAVE_TDM 0
#endif

__global__ __launch_bounds__(NTHR)
void rk4_step_kernel(const float* __restrict__ pin,
                     const float* __restrict__ vin,
                     float* __restrict__ pout) {
  // 5 x 32 KB = 160 KB LDS (<= 320 KB/workgroup on CDNA5): p, v, a1, a2, a3.
  __shared__ float2 smem[5 * TILE * TILE];
  float2* sp = smem;
  float2* sv = smem + 1 * TILE * TILE;
  float2* s1 = smem + 2 * TILE * TILE;
  float2* s2 = smem + 3 * TILE * TILE;
  float2* s3 = smem + 4 * TILE * TILE;

  const int tid = (int)threadIdx.x;
  const int y0 = (int)blockIdx.y * INTR - HALO;   // tile origin incl. halo
  const int x0 = (int)blockIdx.x * INTR - HALO;
  const float2* pin2 = (const float2*)pin;
  const float2* vin2 = (const float2*)vin;

  const bool tile_inside = (y0 >= 0) && (x0 >= 0) &&
                           (y0 + TILE <= SZ) && (x0 + TILE <= SZ);

#if HAVE_TDM
  if (tile_inside) {
    if (tid < 32) {                 // wave 0 issues both DMAs, waits TENSORcnt
      tdm_load_tile(pin2, y0, x0, sp);
      tdm_load_tile(vin2, y0, x0, sv);
      __builtin_amdgcn_s_wait_tensorcnt(0);
    }
  } else
#endif
  {
    // Edge tiles: clamped vector loads (one-sided halo replication; the
    // replicated cells are never consumed by valid outputs).
    __builtin_prefetch(pin2 + (long long)iclampi(y0, 0, SZ - 1) * SZ +
                           iclampi(x0, 0, SZ - 1), 0, 1);
    for (int i = tid; i < TILE * TILE; i += NTHR) {
      const int ly = i >> 6;
      const int lx = i & (TILE - 1);
      const long long g =
          (long long)iclampi(y0 + ly, 0, SZ - 1) * SZ + iclampi(x0 + lx, 0, SZ - 1);
      sp[ly * TILE + lx] = pin2[g];
      sv[ly * TILE + lx] = vin2[g];
    }
  }
  __syncthreads();

  // a1 = p + DT/2 * k1   over [1,63)^2
  stage(s1, sp, sp, sv, 0.5f * DT_F, 1, y0, x0, tid);
  __syncthreads();
  // a2 = p + DT/2 * k2   over [2,62)^2
  stage(s2, s1, sp, sv, 0.5f * DT_F, 2, y0, x0, tid);
  __syncthreads();
  // a3 = p + DT * k3     over [3,61)^2
  stage(s3, s2, sp, sv, DT_F, 3, y0, x0, tid);
  __syncthreads();

  // p_next = p + (DT/6)(k1 + 2k2 + 2k3 + k4)
  //        = p + (a1-p)/3 + 2(a2-p)/3 + (a3-p)/3 + (DT/6) k4
  float2* out2 = (float2*)pout;
  for (int i = tid; i < INTR * INTR; i += NTHR) {
    const int ly = HALO + i / INTR;
    const int lx = HALO + i % INTR;
    const int gy = y0 + ly;
    const int gx = x0 + lx;
    const float2 k4 = rhs_cell(s3, sv, ly, lx, gy, gx);
    const float2 p  = sp[ly * TILE + lx];
    const float2 a1 = s1[ly * TILE + lx];
    const float2 a2 = s2[ly * TILE + lx];
    const float2 a3 = s3[ly * TILE + lx];
    const float c13 = 1.0f / 3.0f, c23 = 2.0f / 3.0f, c6 = DT_F / 6.0f;
    float2 o;
    o.x = p.x + c13 * (a1.x - p.x) + c23 * (a2.x - p.x) + c13 * (a3.x - p.x) + c6 * k4.x;
    o.y = p.y + c13 * (a1.y - p.y) + c23 * (a2.y - p.y) + c13 * (a3.y - p.y) + c6 * k4.y;
    if (gy < SZ && gx < SZ) out2[(long long)gy * SZ + gx] = o;
  }
}

// ---------------------------------------------------------------------------
// Signature probe for V_WMMA_F32_16X16X4_F32 (not launched): frontend accepted
// the 8-arg (bool, v2f, bool, v2f, short, v8f, bool, bool) form last round
// (no diagnostic emitted for it).  This round tests backend ISel; if it
// lowers, the banded-derivative-matrix formulation of the stencil can be
// moved to the matrix pipe (co-executes with VALU).
// A 16x4 f32 = 2 VGPRs, B 4x16 f32 = 2 VGPRs, C/D 16x16 f32 = 8 VGPRs.
// ---------------------------------------------------------------------------
__global__ void wmma_f32_probe(const float* __restrict__ A,
                               const float* __restrict__ B,
                               float* __restrict__ D) {
#if defined(__gfx1250__) && __has_builtin(__builtin_amdgcn_wmma_f32_16x16x4_f32)
  v2f a = ((const v2f*)A)[threadIdx.x];
  v2f b = ((const v2f*)B)[threadIdx.x];
  v8f c = {};
  c = __builtin_amdgcn_wmma_f32_16x16x4_f32(
      /*neg_a=*/false, a, /*neg_b=*/false, b,
      /*c_mod=*/(short)0, c, /*reuse_a=*/false, /*reuse_b=*/false);
  ((v8f*)D)[threadIdx.x] = c;
#else
  (void)A; (void)B; (void)D;
#endif
}

// ---------------------------------------------------------------------------
// Host side: 10 RK4 steps, ping-pong ws -> out so the final step lands in
// d_out.  Inputs are never written.  Needs ws_size >= 4096*4096*2*4 bytes.
// ---------------------------------------------------------------------------
extern "C" void kernel_launch(void* const* d_in, const int* in_sizes, int n_in,
                              void* d_out, int out_size, void* d_ws, size_t ws_size,
                              hipStream_t stream) {
  (void)in_sizes; (void)n_in; (void)out_size; (void)ws_size;
  const float* phi = (const float*)d_in[0];
  const float* v   = (const float*)d_in[1];
  float* out = (float*)d_out;
  float* ws  = (float*)d_ws;

  dim3 grid(NB, NB), block(NTHR);
  const float* src = phi;
  for (int s = 0; s < 10; ++s) {
    float* dst = ((s & 1) == 0) ? ws : out;   // step 9 (odd) -> d_out
    rk4_step_kernel<<<grid, block, 0, stream>>>(src, v, dst);
    src = dst;
  }
}